// GCNBlock_1589137900159
// MI455X (gfx1250) — compile-verified
//
#include <hip/hip_runtime.h>

#define IN_DIM 256
#define OUT_DIM 256
#define BN_EPS 1e-5f
#define BN_NBLK 125

typedef __attribute__((ext_vector_type(2))) float v2f;
typedef __attribute__((ext_vector_type(8))) float v8f;

// ---------------- degree / normalization ----------------

__global__ void deg_init_kernel(unsigned int* __restrict__ deg, int n) {
    int i = blockIdx.x * blockDim.x + threadIdx.x;
    if (i < n) deg[i] = 1u;  // self-loop
}

__global__ void deg_count_kernel(const long long* __restrict__ dst,
                                 unsigned int* __restrict__ deg, int e) {
    int i = blockIdx.x * blockDim.x + threadIdx.x;
    if (i < e) atomicAdd(&deg[(int)dst[i]], 1u);
}

__global__ void dinv_kernel(void* __restrict__ buf, int n) {
    int i = blockIdx.x * blockDim.x + threadIdx.x;
    if (i < n) {
        unsigned int d = ((const unsigned int*)buf)[i];
        ((float*)buf)[i] = rsqrtf((float)d);   // d >= 1 always
    }
}

// ---------------- GEMM: h = x @ W  (FP32 WMMA 16x16x4) ----------------
// Block: 256 threads = 8 waves. Block covers 16 rows x 256 cols.
// Wave w covers columns [w*32, w*32+32) as two 16x16 tiles.

__global__ __launch_bounds__(256) void gemm_wmma_kernel(const float* __restrict__ x,
                                                        const float* __restrict__ W,
                                                        float* __restrict__ h) {
    const int lane  = threadIdx.x & 31;
    const int wave  = threadIdx.x >> 5;
    const int row0  = blockIdx.x * 16;
    const int col0  = wave * 32;
    const int lrow  = lane & 15;      // 0..15
    const int khalf = lane >> 4;      // 0 or 1

    v8f acc0 = {};
    v8f acc1 = {};

    const float* xrow = x + (size_t)(row0 + lrow) * IN_DIM;

    for (int kk = 0; kk < IN_DIM; kk += 4) {
        const int ka = kk + 2 * khalf;
        // A 16x4 layout: lanes 0-15 -> K = kk+0,kk+1 ; lanes 16-31 -> K = kk+2,kk+3
        v2f a;
        a.x = xrow[ka + 0];
        a.y = xrow[ka + 1];
        // B 4x16 layout: VGPR0 -> rows K = kk + 2*khalf ; VGPR1 -> K = kk+1 + 2*khalf
        const float* wp = W + (size_t)ka * OUT_DIM + col0 + lrow;
        v2f b0, b1;
        b0.x = wp[0];
        b0.y = wp[OUT_DIM];
        b1.x = wp[16];
        b1.y = wp[OUT_DIM + 16];
        acc0 = __builtin_amdgcn_wmma_f32_16x16x4_f32(false, a, false, b0,
                                                     (short)0, acc0, false, false);
        acc1 = __builtin_amdgcn_wmma_f32_16x16x4_f32(false, a, false, b1,
                                                     (short)0, acc1, false, false);
    }

    // C/D layout: VGPR r -> row M=r (lanes 0-15), M=r+8 (lanes 16-31)
    float* hp = h + (size_t)(row0 + 8 * khalf) * OUT_DIM + col0 + lrow;
#pragma unroll
    for (int r = 0; r < 8; ++r) {
        hp[(size_t)r * OUT_DIM]      = acc0[r];
        hp[(size_t)r * OUT_DIM + 16] = acc1[r];
    }
}

// Scalar tail for row remainder (unused when n % 16 == 0, kept for safety).
__global__ void gemm_tail_kernel(const float* __restrict__ x, const float* __restrict__ W,
                                 float* __restrict__ h, int rowStart, int n) {
    int idx = blockIdx.x * blockDim.x + threadIdx.x;
    int row = rowStart + (idx >> 8);
    int c   = idx & 255;
    if (row >= n) return;
    const float* xr = x + (size_t)row * IN_DIM;
    float s = 0.f;
    for (int k = 0; k < IN_DIM; ++k) s += xr[k] * W[(size_t)k * OUT_DIM + c];
    h[(size_t)row * OUT_DIM + c] = s;
}

// ---------------- self-loop init: out = h * dinv^2 + b ----------------

__global__ void out_init_kernel(const float* __restrict__ h, const float* __restrict__ dinv,
                                const float* __restrict__ b, float* __restrict__ out,
                                long long total) {
    long long idx = (long long)blockIdx.x * blockDim.x + threadIdx.x;
    if (idx < total) {
        int row = (int)(idx >> 8);
        int c   = (int)(idx & 255);
        float di = dinv[row];
        out[idx] = h[idx] * di * di + b[c];
    }
}

// ---------------- edge scatter: one wave per edge ----------------

__global__ __launch_bounds__(256) void scatter_kernel(const float* __restrict__ h,
                                                      const float* __restrict__ dinv,
                                                      const long long* __restrict__ src,
                                                      const long long* __restrict__ dst,
                                                      float* __restrict__ out, int e) {
    int lane = threadIdx.x & 31;
    int edge = blockIdx.x * (blockDim.x >> 5) + (threadIdx.x >> 5);
    if (edge >= e) return;
    int s = (int)src[edge];
    int d = (int)dst[edge];
    float norm = dinv[s] * dinv[d];
    const float* hp = h + (size_t)s * OUT_DIM;
    float* op = out + (size_t)d * OUT_DIM;
#pragma unroll
    for (int i = 0; i < OUT_DIM / 32; ++i) {
        int c = lane + i * 32;
        unsafeAtomicAdd(&op[c], hp[c] * norm);  // global_atomic_add_f32
    }
}

// ---------------- BatchNorm statistics (deterministic 2-level) ----------------

__global__ __launch_bounds__(256) void bn_partial_kernel(const float* __restrict__ out,
                                                         float* __restrict__ psum,
                                                         float* __restrict__ psq,
                                                         int n, int chunk) {
    int c  = threadIdx.x;              // column 0..255
    int r0 = blockIdx.x * chunk;
    int r1 = r0 + chunk; if (r1 > n) r1 = n;
    float s = 0.f, q = 0.f;
    for (int r = r0; r < r1; ++r) {
        float v = out[(size_t)r * OUT_DIM + c];
        s += v;
        q += v * v;
    }
    psum[blockIdx.x * OUT_DIM + c] = s;
    psq [blockIdx.x * OUT_DIM + c] = q;
}

__global__ void bn_reduce_kernel(const float* __restrict__ psum, const float* __restrict__ psq,
                                 const float* __restrict__ gamma, const float* __restrict__ beta,
                                 float* __restrict__ scale, float* __restrict__ shift,
                                 int n, int nblk) {
    int c = threadIdx.x;
    float s = 0.f, q = 0.f;
    for (int b = 0; b < nblk; ++b) {
        s += psum[b * OUT_DIM + c];
        q += psq [b * OUT_DIM + c];
    }
    float inv_n = 1.0f / (float)n;
    float mean  = s * inv_n;
    float var   = q * inv_n - mean * mean;
    float sc    = gamma[c] * rsqrtf(var + BN_EPS);
    scale[c] = sc;
    shift[c] = beta[c] - mean * sc;
}

__global__ void bn_apply_kernel(float* __restrict__ out, const float* __restrict__ scale,
                                const float* __restrict__ shift, long long total) {
    long long idx = (long long)blockIdx.x * blockDim.x + threadIdx.x;
    if (idx < total) {
        int c = (int)(idx & 255);
        float v = out[idx] * scale[c] + shift[c];
        out[idx] = v > 0.f ? v : 0.f;
    }
}

// ---------------- launch ----------------

extern "C" void kernel_launch(void* const* d_in, const int* in_sizes, int n_in,
                              void* d_out, int out_size, void* d_ws, size_t ws_size,
                              hipStream_t stream) {
    const float*     x     = (const float*)d_in[0];
    const long long* ei    = (const long long*)d_in[1];   // int64 [2, E]
    const float*     W     = (const float*)d_in[2];
    const float*     b     = (const float*)d_in[3];
    const float*     gamma = (const float*)d_in[4];
    const float*     beta  = (const float*)d_in[5];
    float*           out   = (float*)d_out;

    const int n = in_sizes[0] / IN_DIM;
    const int e = in_sizes[1] / 2;
    const long long* src = ei;
    const long long* dst = ei + e;

    // workspace carve-up
    char* ws = (char*)d_ws;
    size_t off = 0;
    float* h = (float*)(ws + off);
    off += (size_t)n * OUT_DIM * sizeof(float);
    off = (off + 255) & ~(size_t)255;
    float* dinv = (float*)(ws + off);           // used as uint deg, then float dinv
    off += (size_t)n * sizeof(float);
    off = (off + 255) & ~(size_t)255;
    float* psum = (float*)(ws + off);
    off += (size_t)BN_NBLK * OUT_DIM * sizeof(float);
    float* psq = (float*)(ws + off);
    off += (size_t)BN_NBLK * OUT_DIM * sizeof(float);
    float* scale = (float*)(ws + off);
    off += OUT_DIM * sizeof(float);
    float* shift = (float*)(ws + off);

    const long long total = (long long)n * OUT_DIM;

    // 1) degrees (exact integer atomics -> deterministic)
    deg_init_kernel<<<(n + 255) / 256, 256, 0, stream>>>((unsigned int*)dinv, n);
    deg_count_kernel<<<(e + 255) / 256, 256, 0, stream>>>(dst, (unsigned int*)dinv, e);
    dinv_kernel<<<(n + 255) / 256, 256, 0, stream>>>((void*)dinv, n);

    // 2) h = x @ W via FP32 WMMA
    int nfull = n / 16;
    if (nfull > 0)
        gemm_wmma_kernel<<<nfull, 256, 0, stream>>>(x, W, h);
    int tail = n - nfull * 16;
    if (tail > 0) {
        int tthreads = tail * OUT_DIM;
        gemm_tail_kernel<<<(tthreads + 255) / 256, 256, 0, stream>>>(x, W, h, nfull * 16, n);
    }

    // 3) out = self-loop term + bias
    out_init_kernel<<<(int)((total + 255) / 256), 256, 0, stream>>>(h, dinv, b, out, total);

    // 4) edge scatter with hardware f32 atomics (one wave per edge)
    scatter_kernel<<<(e + 7) / 8, 256, 0, stream>>>(h, dinv, src, dst, out, e);

    // 5) BatchNorm stats (deterministic two-level reduction) + apply + ReLU
    int chunk = (n + BN_NBLK - 1) / BN_NBLK;
    bn_partial_kernel<<<BN_NBLK, 256, 0, stream>>>(out, psum, psq, n, chunk);
    bn_reduce_kernel<<<1, 256, 0, stream>>>(psum, psq, gamma, beta, scale, shift, n, BN_NBLK);
    bn_apply_kernel<<<(int)((total + 255) / 256), 256, 0, stream>>>(out, scale, shift, total);
}